// Net_31404800868535
// MI455X (gfx1250) — compile-verified
//
#include <hip/hip_runtime.h>
#include <hip/hip_bf16.h>
#include <math.h>

#define NNODES 100000
#define NEDGES 1000000
#define F_IN   33
#define HF     64
#define NH     4
#define NC     2
#define N_ENC  10
#define N_LAY  2
#define EPSV   1e-5f

typedef float v2f __attribute__((ext_vector_type(2)));
typedef float v8f __attribute__((ext_vector_type(8)));

// ---------------------------------------------------------------------------
// WMMA GEMM: C[M x Nout] = A[M x K] @ W[Nout x K]^T (+ bias)
// block = 256 threads = 8 waves; each wave computes one 16x16 output tile.
// grid.x = ceil(M/128), grid.y = Nout/16. Block's 16-column slice of W is
// staged in LDS (row stride padded to 16B so B fragments are ds_load_b64).
// A fragments are unconditional global_load_b64: row index is CLAMPED to
// M-1 instead of predicated, so no exec-mask branches in the hot loop;
// out-of-range rows are simply never stored.
// Uses V_WMMA_F32_16X16X4_F32 (fp32 matrix path; the workload is HBM-bound,
// ~25 GFLOP vs ~6+ GB traffic at 23.3 TB/s, so fp32 keeps reference numerics
// with no throughput penalty).
// ---------------------------------------------------------------------------
template<int K>
__global__ __launch_bounds__(256) void wmma_gemm_nt(
    const float* __restrict__ A, const float* __restrict__ W,
    const float* __restrict__ bias, float* __restrict__ C,
    int M, int Nout)
{
    constexpr int KP = (K + 3) & ~3;          // padded row stride (16B mult.)
    __shared__ float wtile[16 * KP];

    const int ntile = blockIdx.y * 16;
    for (int idx = threadIdx.x; idx < 16 * K; idx += 256) {
        const int n = idx / K, k = idx % K;
        wtile[n * KP + k] = W[(size_t)(ntile + n) * K + k];
    }
    __syncthreads();

    const int wave  = threadIdx.x >> 5;
    const int lane  = threadIdx.x & 31;
    const int hi2   = lane >> 4;              // 0: K offs 0/1, 1: K offs 2/3
    const int l16   = lane & 15;
    const int mtile = blockIdx.x * 128 + wave * 16;

    const int    arow = min(mtile + l16, M - 1);   // clamp: branch-free loads
    const float* Arow = A + (size_t)arow * K;
    const float* Brow = &wtile[l16 * KP];

    v8f acc = {};

    constexpr int kmain = K & ~3;
#pragma unroll 8
    for (int kk = 0; kk < kmain; kk += 4) {
        const int ka = kk + hi2 * 2;
        v2f a, b;
        if constexpr ((K & 3) == 0) {
            a = *(const v2f*)(Arow + ka);     // 8B-aligned global_load_b64
        } else {
            a.x = Arow[ka];                   // K=33: rows only 4B-aligned
            a.y = Arow[ka + 1];
        }
        b = *(const v2f*)(Brow + ka);         // 8B-aligned ds_load_b64
        acc = __builtin_amdgcn_wmma_f32_16x16x4_f32(
            /*neg_a=*/false, a, /*neg_b=*/false, b,
            /*c_mod=*/(short)0, acc, /*reuse_a=*/false, /*reuse_b=*/false);
    }
    if constexpr (kmain < K) {                // scalar remainder (K = 33)
        for (int k = kmain; k < K; ++k) {
            const float bv = Brow[k];
#pragma unroll
            for (int vj = 0; vj < 8; ++vj) {
                const int m = min(mtile + vj + hi2 * 8, M - 1);
                acc[vj] += A[(size_t)m * K + k] * bv;
            }
        }
    }

    const float bb = bias ? bias[ntile + l16] : 0.0f;
#pragma unroll
    for (int vj = 0; vj < 8; ++vj) {
        const int m = mtile + vj + hi2 * 8;
        if (m < M) C[(size_t)m * Nout + ntile + l16] = acc[vj] + bb;
    }
}

// ---------------------------------------------------------------------------
// BatchNorm column statistics: one block per feature column (64 blocks).
// stats[c] = mean, stats[64+c] = rsqrt(var + eps)
// ---------------------------------------------------------------------------
__global__ __launch_bounds__(256) void bn_stats_kernel(
    const float* __restrict__ h, float* __restrict__ stats, int M)
{
    const int col = blockIdx.x;
    float s = 0.0f, s2 = 0.0f;
    for (int i = threadIdx.x; i < M; i += 256) {
        const float v = h[(size_t)i * HF + col];
        s += v; s2 += v * v;
    }
    __shared__ float ls[256], ls2[256];
    ls[threadIdx.x] = s; ls2[threadIdx.x] = s2;
    __syncthreads();
    for (int off = 128; off > 0; off >>= 1) {
        if (threadIdx.x < off) {
            ls[threadIdx.x]  += ls[threadIdx.x + off];
            ls2[threadIdx.x] += ls2[threadIdx.x + off];
        }
        __syncthreads();
    }
    if (threadIdx.x == 0) {
        const float m   = ls[0] / (float)M;
        const float var = ls2[0] / (float)M - m * m;   // biased variance
        stats[col]      = m;
        stats[HF + col] = rsqrtf(var + EPSV);
    }
}

// ---------------------------------------------------------------------------
// BN apply + LeakyReLU, elementwise over N*HF
// ---------------------------------------------------------------------------
__global__ __launch_bounds__(256) void bn_lrelu_kernel(
    const float* __restrict__ x, const float* __restrict__ stats,
    const float* __restrict__ g, const float* __restrict__ b,
    float* __restrict__ out, int total)
{
    for (int idx = blockIdx.x * 256 + threadIdx.x; idx < total;
         idx += gridDim.x * 256) {
        const int c = idx & (HF - 1);
        const float v = g[c] * (x[idx] - stats[c]) * stats[HF + c] + b[c];
        out[idx] = v > 0.0f ? v : 0.01f * v;
    }
}

// ---------------------------------------------------------------------------
// zero fill
// ---------------------------------------------------------------------------
__global__ __launch_bounds__(256) void fill_zero_kernel(float* __restrict__ p, int total)
{
    for (int idx = blockIdx.x * 256 + threadIdx.x; idx < total;
         idx += gridDim.x * 256) p[idx] = 0.0f;
}

// ---------------------------------------------------------------------------
// Edge attention: one wave32 per edge.
//   score[h] = <Q[src,h,:], K[dst,h,:]> * scale   (64-dim dot, float2/lane)
//   attn     = softmax over the 4 heads (faithful to reference: heads dim)
//   hnew[dst,h,:] += V[src,h,:] * attn[h]         (global float atomics)
// ---------------------------------------------------------------------------
__global__ __launch_bounds__(256) void edge_attn_kernel(
    const float* __restrict__ Q, const float* __restrict__ Km,
    const float* __restrict__ V, const int* __restrict__ src,
    const int* __restrict__ dst, float* __restrict__ hnew,
    int E, float scale)
{
    const int lane   = threadIdx.x & 31;
    const int wv     = (blockIdx.x * 256 + threadIdx.x) >> 5;
    const int nwaves = (gridDim.x * 256) >> 5;

    for (int e = wv; e < E; e += nwaves) {
        const int s = src[e];
        const int d = dst[e];
        const float* qp = Q  + (size_t)s * (NH * HF);
        const float* kp = Km + (size_t)d * (NH * HF);

        float sc[NH];
#pragma unroll
        for (int h = 0; h < NH; ++h) {
            const float2 q = *(const float2*)(qp + h * HF + lane * 2);
            const float2 k = *(const float2*)(kp + h * HF + lane * 2);
            float p = q.x * k.x + q.y * k.y;
#pragma unroll
            for (int off = 16; off > 0; off >>= 1) p += __shfl_xor(p, off, 32);
            sc[h] = p * scale;   // every lane holds the full dot now
        }
        // softmax over heads (replicated in every lane)
        float mx = fmaxf(fmaxf(sc[0], sc[1]), fmaxf(sc[2], sc[3]));
        float es[NH], ssum = 0.0f;
#pragma unroll
        for (int h = 0; h < NH; ++h) { es[h] = __expf(sc[h] - mx); ssum += es[h]; }
        const float inv = 1.0f / ssum;

        const float* vp = V    + (size_t)s * (NH * HF);
        float*       op = hnew + (size_t)d * (NH * HF);
#pragma unroll
        for (int h = 0; h < NH; ++h) {
            const float a = es[h] * inv;
            const float2 v = *(const float2*)(vp + h * HF + lane * 2);
            atomicAdd(op + h * HF + lane * 2,     v.x * a);
            atomicAdd(op + h * HF + lane * 2 + 1, v.y * a);
        }
    }
}

// ---------------------------------------------------------------------------
// LayerNorm over last dim (HF=64) with residual add: out = LN(x + res)
// one wave32 per row, 2 elements per lane
// ---------------------------------------------------------------------------
__global__ __launch_bounds__(256) void ln_add_kernel(
    const float* __restrict__ x, const float* __restrict__ res,
    const float* __restrict__ g, const float* __restrict__ b,
    float* __restrict__ out, int M)
{
    const int lane   = threadIdx.x & 31;
    const int wv     = (blockIdx.x * 256 + threadIdx.x) >> 5;
    const int nwaves = (gridDim.x * 256) >> 5;

    for (int i = wv; i < M; i += nwaves) {
        const size_t base = (size_t)i * HF + lane * 2;
        float v0 = x[base]     + res[base];
        float v1 = x[base + 1] + res[base + 1];
        float s = v0 + v1;
#pragma unroll
        for (int off = 16; off > 0; off >>= 1) s += __shfl_xor(s, off, 32);
        const float mean = s * (1.0f / HF);
        const float d0 = v0 - mean, d1 = v1 - mean;
        float vs = d0 * d0 + d1 * d1;
#pragma unroll
        for (int off = 16; off > 0; off >>= 1) vs += __shfl_xor(vs, off, 32);
        const float invstd = rsqrtf(vs * (1.0f / HF) + EPSV);
        out[base]     = g[lane * 2]     * d0 * invstd + b[lane * 2];
        out[base + 1] = g[lane * 2 + 1] * d1 * invstd + b[lane * 2 + 1];
    }
}

// ---------------------------------------------------------------------------
// Final FC: out[i,c] = <h[i,:], fcW[c,:]> + fcb[c]   (NC=2)
// ---------------------------------------------------------------------------
__global__ __launch_bounds__(256) void fc_kernel(
    const float* __restrict__ h, const float* __restrict__ fcW,
    const float* __restrict__ fcb, float* __restrict__ out, int M)
{
    const int i = blockIdx.x * 256 + threadIdx.x;
    if (i >= M) return;
    const float* hr = h + (size_t)i * HF;
#pragma unroll
    for (int c = 0; c < NC; ++c) {
        float s = fcb[c];
        for (int k = 0; k < HF; ++k) s += hr[k] * fcW[c * HF + k];
        out[(size_t)i * NC + c] = s;
    }
}

// ---------------------------------------------------------------------------
// host-side launch
// ---------------------------------------------------------------------------
extern "C" void kernel_launch(void* const* d_in, const int* in_sizes, int n_in,
                              void* d_out, int out_size, void* d_ws, size_t ws_size,
                              hipStream_t stream) {
    // input order from setup_inputs(): feat, src, dst, enc_W[10], enc_b[10],
    // bn_g[10], bn_b[10], Wq, Wk, Wv, Wo, ln_g, ln_b, fc_W, fc_b
    const float* feat = (const float*)d_in[0];
    const int*   src  = (const int*)d_in[1];
    const int*   dst  = (const int*)d_in[2];
    const float* encW[N_ENC]; const float* encB[N_ENC];
    const float* bnG[N_ENC];  const float* bnB[N_ENC];
    for (int i = 0; i < N_ENC; ++i) {
        encW[i] = (const float*)d_in[3 + i];
        encB[i] = (const float*)d_in[13 + i];
        bnG[i]  = (const float*)d_in[23 + i];
        bnB[i]  = (const float*)d_in[33 + i];
    }
    const float* Wq  = (const float*)d_in[43];
    const float* Wk  = (const float*)d_in[44];
    const float* Wv  = (const float*)d_in[45];
    const float* Wo  = (const float*)d_in[46];
    const float* lnG = (const float*)d_in[47];
    const float* lnB = (const float*)d_in[48];
    const float* fcW = (const float*)d_in[49];
    const float* fcB = (const float*)d_in[50];

    // workspace layout (floats)
    float* ws    = (float*)d_ws;
    float* h0    = ws;                                   // N x 64
    float* h1    = h0 + (size_t)NNODES * HF;             // N x 64
    float* stats = h1 + (size_t)NNODES * HF;             // 256 (mean | invstd)
    float* Qb    = stats + 256;                          // N x 256
    float* Kb    = Qb + (size_t)NNODES * NH * HF;        // N x 256
    float* Vb    = Kb + (size_t)NNODES * NH * HF;        // N x 256
    float* hnew  = Vb + (size_t)NNODES * NH * HF;        // N x 256

    const dim3 blk(256);
    const dim3 gemm_grid64((NNODES + 127) / 128, HF / 16);         // Nout=64
    const dim3 gemm_grid256((NNODES + 127) / 128, (NH * HF) / 16); // Nout=256
    const int ew_grid = 4096;    // grid-stride
    const int fc_grid = (NNODES + 255) / 256;

    // ------------------ encoder: 10x (GEMM + BN + LeakyReLU) ---------------
    wmma_gemm_nt<F_IN><<<gemm_grid64, blk, 0, stream>>>(feat, encW[0], encB[0],
                                                        h1, NNODES, HF);
    bn_stats_kernel<<<HF, blk, 0, stream>>>(h1, stats, NNODES);
    bn_lrelu_kernel<<<2048, blk, 0, stream>>>(h1, stats, bnG[0], bnB[0],
                                              h0, NNODES * HF);
    for (int i = 1; i < N_ENC; ++i) {
        wmma_gemm_nt<HF><<<gemm_grid64, blk, 0, stream>>>(h0, encW[i], encB[i],
                                                          h1, NNODES, HF);
        bn_stats_kernel<<<HF, blk, 0, stream>>>(h1, stats, NNODES);
        bn_lrelu_kernel<<<2048, blk, 0, stream>>>(h1, stats, bnG[i], bnB[i],
                                                  h0, NNODES * HF);
    }

    // ------------------ 2 graph-transformer layers -------------------------
    const float scale = 0.125f;   // 1/sqrt(64)
    for (int l = 0; l < N_LAY; ++l) {
        const size_t woff = (size_t)l * (NH * HF) * HF;   // 256*64 per layer
        wmma_gemm_nt<HF><<<gemm_grid256, blk, 0, stream>>>(h0, Wq + woff, nullptr,
                                                           Qb, NNODES, NH * HF);
        wmma_gemm_nt<HF><<<gemm_grid256, blk, 0, stream>>>(h0, Wk + woff, nullptr,
                                                           Kb, NNODES, NH * HF);
        wmma_gemm_nt<HF><<<gemm_grid256, blk, 0, stream>>>(h0, Wv + woff, nullptr,
                                                           Vb, NNODES, NH * HF);
        fill_zero_kernel<<<2048, blk, 0, stream>>>(hnew, NNODES * NH * HF);
        edge_attn_kernel<<<ew_grid, blk, 0, stream>>>(Qb, Kb, Vb, src, dst,
                                                      hnew, NEDGES, scale);
        // Wo: [64 x 256] -> h1 = hnew @ Wo^T
        wmma_gemm_nt<NH * HF><<<gemm_grid64, blk, 0, stream>>>(hnew, Wo + woff,
                                                               nullptr, h1,
                                                               NNODES, HF);
        ln_add_kernel<<<2048, blk, 0, stream>>>(h1, h0, lnG + l * HF,
                                                lnB + l * HF, h0, NNODES);
    }

    // ------------------ final FC ------------------------------------------
    fc_kernel<<<fc_grid, blk, 0, stream>>>(h0, fcW, fcB, (float*)d_out, NNODES);
}